// Megnet_NodeModel_36301063586429
// MI455X (gfx1250) — compile-verified
//
#include <hip/hip_runtime.h>

typedef float v2f __attribute__((ext_vector_type(2)));
typedef float v8f __attribute__((ext_vector_type(8)));
typedef int   v4i __attribute__((vector_size(16)));   // matches builtin param type

#define NN    100000
#define EE    1600000
#define DIMM  128
#define KCAT  (3 * DIMM)   // 384
#define BN_EPSF 1e-5f

// ---------------------------------------------------------------------------
// CDNA5 async global->LDS staging (ASYNCcnt path), with sync fallback.
// Builtin signature (from hipcc diagnostic): (v4i as1* src, v4i as3* dst,
// imm int offset, imm int cpol).
// ---------------------------------------------------------------------------
#if __has_builtin(__builtin_amdgcn_global_load_async_to_lds_b128) && \
    __has_builtin(__builtin_amdgcn_s_wait_asynccnt)
#define USE_ASYNC_LDS 1
#else
#define USE_ASYNC_LDS 0
#endif

__device__ __forceinline__ void stage_b128(const float* g, float* l) {
#if USE_ASYNC_LDS
  __builtin_amdgcn_global_load_async_to_lds_b128(
      (__attribute__((address_space(1))) v4i*)g,
      (__attribute__((address_space(3))) v4i*)l,
      0, 0);
#else
  *(float4*)l = *(const float4*)g;
#endif
}

__device__ __forceinline__ void stage_wait() {
#if USE_ASYNC_LDS
  __builtin_amdgcn_s_wait_asynccnt(0);
#endif
  __syncthreads();
}

// ---------------------------------------------------------------------------
// Zero workspace accumulators (BN stats, degree histogram, alloc counter).
// ---------------------------------------------------------------------------
__global__ void zero_kernel(float* __restrict__ p, long n) {
  long i = (long)blockIdx.x * blockDim.x + threadIdx.x;
  long stride = (long)gridDim.x * blockDim.x;
  for (; i < n; i += stride) p[i] = 0.0f;
}

// ---------------------------------------------------------------------------
// Binned scatter_mean, step 1: per-node degree histogram (int atomics only).
// ---------------------------------------------------------------------------
__global__ void deg_kernel(const int* __restrict__ src, int* __restrict__ deg) {
  long e = (long)blockIdx.x * blockDim.x + threadIdx.x;
  if (e >= EE) return;
  atomicAdd(&deg[src[e]], 1);
}

// Step 2: allocate a disjoint bin per node (order irrelevant -> no scan).
__global__ void alloc_kernel(const int* __restrict__ deg, int* __restrict__ total,
                             int* __restrict__ base, int* __restrict__ cursor) {
  long n = (long)blockIdx.x * blockDim.x + threadIdx.x;
  if (n >= NN) return;
  int b = atomicAdd(total, deg[n]);
  base[n]   = b;
  cursor[n] = b;
}

// Step 3: bin edge ids by src node.
__global__ void bin_kernel(const int* __restrict__ src, int* __restrict__ cursor,
                           int* __restrict__ perm) {
  long e = (long)blockIdx.x * blockDim.x + threadIdx.x;
  if (e >= EE) return;
  int pos = atomicAdd(&cursor[src[e]], 1);
  perm[pos] = (int)e;
}

// Step 4: per-node gather-sum (no fp atomics). One 128-thread block per node;
// each 512B edge row is read coalesced across the 128 lanes.
__global__ void gather_mean_kernel(const float* __restrict__ edge_attr,
                                   const int* __restrict__ perm,
                                   const int* __restrict__ base,
                                   const int* __restrict__ deg,
                                   float* __restrict__ ve) {
  int n   = blockIdx.x;
  int col = threadIdx.x;
  int s = base[n];
  int d = deg[n];
  float acc = 0.0f;
  for (int j = 0; j < d; ++j) {
    long e = perm[s + j];
    acc += edge_attr[e * DIMM + col];
  }
  ve[(long)n * DIMM + col] = acc / fmaxf((float)d, 1.0f);
}

// ---------------------------------------------------------------------------
// Fold input-side BatchNorm into weights/bias:
//   (h*s + t) @ W^T + b  ==  h @ (W*s)^T + (b + W @ t)
// Grid: 128 blocks (output n), 128 threads (input k).
// ---------------------------------------------------------------------------
__global__ void foldbn_kernel(const float* __restrict__ W, const float* __restrict__ b,
                              const float* __restrict__ colsum,
                              const float* __restrict__ colsumsq,
                              const float* __restrict__ gamma,
                              const float* __restrict__ beta,
                              float* __restrict__ Wf, float* __restrict__ bf) {
  __shared__ float red[DIMM];
  int n = blockIdx.x;
  int k = threadIdx.x;
  float mu   = colsum[k] * (1.0f / NN);
  float var  = colsumsq[k] * (1.0f / NN) - mu * mu;
  float rstd = rsqrtf(var + BN_EPSF);
  float s = gamma[k] * rstd;
  float t = beta[k] - mu * s;
  float w = W[n * DIMM + k];
  Wf[n * DIMM + k] = w * s;
  red[k] = w * t;
  __syncthreads();
  for (int off = 64; off > 0; off >>= 1) {
    if (k < off) red[k] += red[k + off];
    __syncthreads();
  }
  if (k == 0) bf[n] = b[n] + red[0];
}

// ---------------------------------------------------------------------------
// Fused GEMM + bias + ReLU on V_WMMA_F32_16X16X4_F32.
// Block = 256 threads (8 wave32) -> 32 output rows x 128 output cols.
// Each wave: one 16-col tile, two 16x16 M-tiles (B fragment reused).
// A staged to LDS via async global->LDS b128 (pure copy: BN pre-folded).
// LDS rows padded to K+4 dwords so 16 A-lane reads hit distinct banks.
// ---------------------------------------------------------------------------
template <int MODE, int K>
__global__ __launch_bounds__(256) void gemm_relu_kernel(
    const float* __restrict__ A0,     // MODE0: x [N,128]; MODE1: h_prev [N,128]
    const float* __restrict__ ve,     // MODE0 only [N,128]
    const float* __restrict__ u,      // MODE0 only [G,128]
    const int*   __restrict__ batch,  // MODE0 only [N]
    const float* __restrict__ W,      // [128,K] (BN-folded for MODE1)
    const float* __restrict__ bias,   // [128]   (BN-folded for MODE1)
    float* __restrict__ out)          // [N,128] raw relu activations
{
  constexpr int KP = K + 4;           // bank-conflict padding (K % 64 == 0)
  __shared__ float ldsA[32 * KP];

  const int  tid  = threadIdx.x;
  const long row0 = (long)blockIdx.x * 32;

  // Stage the 32 x K A-tile (concat gather for MODE0, straight copy for MODE1).
  for (int i = tid; i < 32 * (K / 4); i += 256) {
    int r = i / (K / 4);
    int c = (i % (K / 4)) * 4;
    long row = row0 + r;
    const float* gsrc;
    if constexpr (MODE == 0) {
      if (c < DIMM)            gsrc = A0 + row * DIMM + c;
      else if (c < 2 * DIMM)   gsrc = ve + row * DIMM + (c - DIMM);
      else                     gsrc = u + (long)batch[row] * DIMM + (c - 2 * DIMM);
    } else {
      gsrc = A0 + row * DIMM + c;
    }
    stage_b128(gsrc, &ldsA[r * KP + c]);
  }
  stage_wait();

  const int wave = tid >> 5;
  const int lane = tid & 31;
  const int hh   = lane >> 4;   // K-pair half (lanes 16-31 supply K+2,K+3)
  const int l16  = lane & 15;
  const int n0   = wave * 16;

  v8f acc0 = {0.f, 0.f, 0.f, 0.f, 0.f, 0.f, 0.f, 0.f};
  v8f acc1 = {0.f, 0.f, 0.f, 0.f, 0.f, 0.f, 0.f, 0.f};
  const float* wp    = W + (long)(n0 + l16) * K;  // B[k][n] = W[n][k]
  const float* arow0 = ldsA + l16 * KP;
  const float* arow1 = ldsA + (16 + l16) * KP;

  for (int k = 0; k < K; k += 4) {
    int kk = k + 2 * hh;
    v2f a0, a1, b;
    a0.x = arow0[kk]; a0.y = arow0[kk + 1];
    a1.x = arow1[kk]; a1.y = arow1[kk + 1];
    b.x  = wp[kk];    b.y  = wp[kk + 1];
    acc0 = __builtin_amdgcn_wmma_f32_16x16x4_f32(false, a0, false, b,
                                                 (short)0, acc0, false, false);
    acc1 = __builtin_amdgcn_wmma_f32_16x16x4_f32(false, a1, false, b,
                                                 (short)0, acc1, false, false);
  }

  const float bb  = bias[n0 + l16];
  const long  col = n0 + l16;
#pragma unroll
  for (int r = 0; r < 8; ++r) {
    int m = r + 8 * hh;  // C/D layout: VGPR r, half hh -> row m
    out[(row0 + m)      * DIMM + col] = fmaxf(acc0[r] + bb, 0.0f);
    out[(row0 + 16 + m) * DIMM + col] = fmaxf(acc1[r] + bb, 0.0f);
  }
}

// ---------------------------------------------------------------------------
// Column-wise sum / sum-of-squares for training-mode BN (51MB sweep ~2us).
// ---------------------------------------------------------------------------
__global__ void colstats_kernel(const float* __restrict__ h,
                                float* __restrict__ colsum,
                                float* __restrict__ colsumsq) {
  int col = threadIdx.x;  // 0..127, coalesced
  float s = 0.0f, s2 = 0.0f;
  for (long row = blockIdx.x; row < NN; row += gridDim.x) {
    float v = h[row * DIMM + col];
    s += v;
    s2 += v * v;
  }
  atomicAdd(colsum + col, s);
  atomicAdd(colsumsq + col, s2);
}

// ---------------------------------------------------------------------------
// Final BN apply (layer 2) -> d_out.
// ---------------------------------------------------------------------------
__global__ void bnapply_kernel(const float* __restrict__ h,
                               const float* __restrict__ colsum,
                               const float* __restrict__ colsumsq,
                               const float* __restrict__ gamma,
                               const float* __restrict__ beta,
                               float* __restrict__ out) {
  long i = (long)blockIdx.x * blockDim.x + threadIdx.x;
  if (i >= (long)NN * DIMM) return;
  int c = (int)(i % DIMM);
  float mu   = colsum[c] * (1.0f / NN);
  float var  = colsumsq[c] * (1.0f / NN) - mu * mu;
  float rstd = rsqrtf(var + BN_EPSF);
  out[i] = (h[i] - mu) * rstd * gamma[c] + beta[c];
}

// ---------------------------------------------------------------------------
extern "C" void kernel_launch(void* const* d_in, const int* in_sizes, int n_in,
                              void* d_out, int out_size, void* d_ws, size_t ws_size,
                              hipStream_t stream) {
  (void)in_sizes; (void)n_in; (void)out_size; (void)ws_size;

  const float* x         = (const float*)d_in[0];
  const float* edge_attr = (const float*)d_in[1];
  const float* u         = (const float*)d_in[2];
  const int*   edge_idx  = (const int*)d_in[3];   // [2,E]; row 0 = src
  const int*   batch     = (const int*)d_in[4];
  const float* W0 = (const float*)d_in[5];
  const float* b0 = (const float*)d_in[6];
  const float* W1 = (const float*)d_in[7];
  const float* b1 = (const float*)d_in[8];
  const float* W2 = (const float*)d_in[9];
  const float* b2 = (const float*)d_in[10];
  const float* g0 = (const float*)d_in[11];
  const float* bt0 = (const float*)d_in[12];
  const float* g1 = (const float*)d_in[13];
  const float* bt1 = (const float*)d_in[14];
  const float* g2 = (const float*)d_in[15];
  const float* bt2 = (const float*)d_in[16];

  float* out = (float*)d_out;
  float* ws  = (float*)d_ws;

  // Workspace layout:
  float* ve    = ws;                        // [N*128] v_e, later reused as h2
  float* B1    = ve + (long)NN * DIMM;      // [N*128] h1
  float* Wf1   = B1 + (long)NN * DIMM;      // [128*128] BN-folded W1
  float* bf1   = Wf1 + DIMM * DIMM;         // [128]
  float* Wf2   = bf1 + DIMM;                // [128*128] BN-folded W2
  float* bf2   = Wf2 + DIMM * DIMM;         // [128]
  float* stats = bf2 + DIMM;                // [6*128] {sum0,sq0,sum1,sq1,sum2,sq2}
  int*   deg    = (int*)(stats + 6 * DIMM); // [N]
  int*   total  = deg + NN;                 // [1]
  int*   base   = total + 1;                // [N]
  int*   cursor = base + NN;                // [N]
  int*   perm   = cursor + NN;              // [E]

  // 1) zero stats + deg + total (contiguous; int 0 == float 0.0 bit pattern)
  zero_kernel<<<512, 256, 0, stream>>>(stats, 6 * DIMM + NN + 1);

  // 2) binned scatter_mean (no fp32 atomics)
  deg_kernel  <<<(EE + 255) / 256, 256, 0, stream>>>(edge_idx, deg);
  alloc_kernel<<<(NN + 255) / 256, 256, 0, stream>>>(deg, total, base, cursor);
  bin_kernel  <<<(EE + 255) / 256, 256, 0, stream>>>(edge_idx, cursor, perm);
  gather_mean_kernel<<<NN, DIMM, 0, stream>>>(edge_attr, perm, base, deg, ve);

  // 3) layer 0: concat -> GEMM(384->128) + ReLU -> h0 in d_out
  gemm_relu_kernel<0, KCAT><<<NN / 32, 256, 0, stream>>>(
      x, ve, u, batch, W0, b0, out);
  colstats_kernel<<<512, DIMM, 0, stream>>>(out, stats + 0, stats + 128);

  // 4) layer 1: fold BN0 into W1/b1, then GEMM(128->128) + ReLU -> h1 in B1
  foldbn_kernel<<<DIMM, DIMM, 0, stream>>>(W1, b1, stats + 0, stats + 128,
                                           g0, bt0, Wf1, bf1);
  gemm_relu_kernel<1, DIMM><<<NN / 32, 256, 0, stream>>>(
      out, nullptr, nullptr, nullptr, Wf1, bf1, B1);
  colstats_kernel<<<512, DIMM, 0, stream>>>(B1, stats + 256, stats + 384);

  // 5) layer 2: fold BN1 into W2/b2, GEMM -> h2 in ve (v_e is dead)
  foldbn_kernel<<<DIMM, DIMM, 0, stream>>>(W2, b2, stats + 256, stats + 384,
                                           g1, bt1, Wf2, bf2);
  gemm_relu_kernel<1, DIMM><<<NN / 32, 256, 0, stream>>>(
      B1, nullptr, nullptr, nullptr, Wf2, bf2, ve);
  colstats_kernel<<<512, DIMM, 0, stream>>>(ve, stats + 512, stats + 640);

  // 6) final BN2 apply -> d_out
  long tot = (long)NN * DIMM;
  bnapply_kernel<<<(unsigned)((tot + 255) / 256), 256, 0, stream>>>(
      ve, stats + 512, stats + 640, g2, bt2, out);
}